// CTAM_SSCL_Loss_45311904973350
// MI455X (gfx1250) — compile-verified
//
#include <hip/hip_runtime.h>
#include <stdint.h>

#define B_ 64
#define N_ 65536
#define D_ 2048
#define SPLITS 16
#define TPB 512
#define CHUNK (N_ / SPLITS)        // 4096 elements per block
#define INV_T (1.0f / 0.07f)       // TEMP = BASE_TEMP = 0.07 -> scale factor 1

typedef float v2f __attribute__((ext_vector_type(2)));
typedef float v8f __attribute__((ext_vector_type(8)));
typedef unsigned int v4u __attribute__((ext_vector_type(4)));
typedef int v4i __attribute__((ext_vector_type(4)));
typedef int v8i __attribute__((ext_vector_type(8)));

__device__ __forceinline__ void combine_ms(float& m, float& s, float m2, float s2) {
    float nm = fmaxf(m, m2);
    s = s * __expf(m - nm) + s2 * __expf(m2 - nm);   // sentinel -1e30 underflows to 0
    m = nm;
}
__device__ __forceinline__ void combine_min(float& v, int& i, float v2, int i2) {
    if (v2 < v || (v2 == v && i2 < i)) { v = v2; i = i2; }
}

// ---------------------------------------------------------------------------
// Kernel 1: per-(sample, split) masked online-softmax partials over CHUNK elems
// ---------------------------------------------------------------------------
__global__ __launch_bounds__(TPB) void partial_kernel(
    const float* __restrict__ logits, const int* __restrict__ memCID,
    const int* __restrict__ memTID, const int* __restrict__ camids,
    const int* __restrict__ trackids,
    float* __restrict__ pm, float* __restrict__ ps, float* __restrict__ pn,
    float* __restrict__ pp, float* __restrict__ pv, int* __restrict__ pi) {
    const int split = blockIdx.x;
    const int b = blockIdx.y;
    const int camid = camids[b];
    const int trackid = trackids[b];
    const float* row = logits + (size_t)b * N_;

    float m = -1e30f, s = 0.f, np = 0.f, psum = 0.f, minv = 1e30f;
    int mini = 0x7fffffff;

    const int n0 = split * CHUNK + threadIdx.x * 4;   // 16B aligned
#pragma unroll
    for (int j = 0; j < CHUNK / (TPB * 4); ++j) {     // 2 iterations
        const int n = n0 + j * (TPB * 4);
        float4 lg = *(const float4*)(row + n);
        int4 c4 = *(const int4*)(memCID + n);
        int4 t4 = *(const int4*)(memTID + n);
        float lv[4] = {lg.x, lg.y, lg.z, lg.w};
        int cv[4] = {c4.x, c4.y, c4.z, c4.w};
        int tv[4] = {t4.x, t4.y, t4.z, t4.w};
#pragma unroll
        for (int k = 0; k < 4; ++k) {
            if (cv[k] == camid) {
                float adc = lv[k] * INV_T;
                float nm = fmaxf(m, adc);
                s = s * __expf(m - nm) + __expf(adc - nm);
                m = nm;
                if (tv[k] == trackid) {
                    np += 1.f;
                    psum += adc;
                    int idx = n + k;
                    if (lv[k] < minv || (lv[k] == minv && idx < mini)) {
                        minv = lv[k]; mini = idx;
                    }
                }
            }
        }
    }

    // wave32 xor-shuffle reduction
#pragma unroll
    for (int off = 16; off > 0; off >>= 1) {
        combine_ms(m, s, __shfl_xor(m, off), __shfl_xor(s, off));
        np += __shfl_xor(np, off);
        psum += __shfl_xor(psum, off);
        combine_min(minv, mini, __shfl_xor(minv, off), __shfl_xor(mini, off));
    }

    __shared__ float lm[TPB / 32], lss[TPB / 32], lnp[TPB / 32], lps[TPB / 32], lmv[TPB / 32];
    __shared__ int lmi[TPB / 32];
    const int wave = threadIdx.x >> 5, lane = threadIdx.x & 31;
    if (lane == 0) {
        lm[wave] = m; lss[wave] = s; lnp[wave] = np;
        lps[wave] = psum; lmv[wave] = minv; lmi[wave] = mini;
    }
    __syncthreads();
    if (wave == 0) {
        const bool ok = lane < (TPB / 32);
        m = ok ? lm[lane] : -1e30f;   s = ok ? lss[lane] : 0.f;
        np = ok ? lnp[lane] : 0.f;    psum = ok ? lps[lane] : 0.f;
        minv = ok ? lmv[lane] : 1e30f; mini = ok ? lmi[lane] : 0x7fffffff;
#pragma unroll
        for (int off = 8; off > 0; off >>= 1) {
            combine_ms(m, s, __shfl_xor(m, off), __shfl_xor(s, off));
            np += __shfl_xor(np, off);
            psum += __shfl_xor(psum, off);
            combine_min(minv, mini, __shfl_xor(minv, off), __shfl_xor(mini, off));
        }
        if (lane == 0) {
            const int p = b * SPLITS + split;
            pm[p] = m; ps[p] = s; pn[p] = np; pp[p] = psum; pv[p] = minv; pi[p] = mini;
        }
    }
}

// ---------------------------------------------------------------------------
// Kernel 2: fold the SPLITS partials per sample -> per-sample loss + hard idx
// ---------------------------------------------------------------------------
__global__ __launch_bounds__(B_) void combine_kernel(
    const float* __restrict__ pm, const float* __restrict__ ps,
    const float* __restrict__ pn, const float* __restrict__ pp,
    const float* __restrict__ pv, const int* __restrict__ pi,
    float* __restrict__ lossb, int* __restrict__ hidx) {
    const int b = threadIdx.x;
    float m = -1e30f, s = 0.f, np = 0.f, psum = 0.f, minv = 1e30f;
    int mini = 0x7fffffff;
#pragma unroll
    for (int k = 0; k < SPLITS; ++k) {
        const int p = b * SPLITS + k;
        combine_ms(m, s, pm[p], ps[p]);
        np += pn[p];
        psum += pp[p];
        combine_min(minv, mini, pv[p], pi[p]);
    }
    const float lse = __logf(s);
    // logprob_sum = sum_pos(adc) - num_pos*(m + lse);  loss = -logprob_sum/num_pos
    lossb[b] = -(psum - np * (m + lse)) / np;
    hidx[b] = mini;
}

// ---------------------------------------------------------------------------
// Kernel 3: mean of 64 losses via v_wmma_f32_16x16x4_f32 (A=losses 16x4, B=ones)
// ---------------------------------------------------------------------------
__global__ __launch_bounds__(32) void mean_kernel(const float* __restrict__ lossb,
                                                  float* __restrict__ out) {
    const int lane = threadIdx.x;
#if __has_builtin(__builtin_amdgcn_wmma_f32_16x16x4_f32)
    // A[M][K] = loss[4*M+K]: VGPR0 = {K=0 (lanes 0-15), K=2 (lanes 16-31)}, VGPR1 = {K=1, K=3}
    const int mrow = lane & 15;
    const int base = 4 * mrow + ((lane >> 4) << 1);
    v2f a; a.x = lossb[base]; a.y = lossb[base + 1];
    v2f ones; ones.x = 1.0f; ones.y = 1.0f;
    v8f c = {};
    c = __builtin_amdgcn_wmma_f32_16x16x4_f32(false, a, false, ones, (short)0, c, false, false);
    // every column of D holds the 16 row-sums; lane j has rows 0-7 (j<16) / 8-15 (j>=16)
    float ssum = c[0] + c[1] + c[2] + c[3] + c[4] + c[5] + c[6] + c[7];
    ssum += __shfl_xor(ssum, 16);
    if (lane == 0) out[0] = ssum * (1.0f / (float)B_);
#else
    float v = lossb[lane] + lossb[lane + 32];
#pragma unroll
    for (int off = 16; off > 0; off >>= 1) v += __shfl_xor(v, off);
    if (lane == 0) out[0] = v * (1.0f / (float)B_);
#endif
}

// ---------------------------------------------------------------------------
// Kernel 4: hard-positive row gather via the Tensor Data Mover (TDM)
//   tensor_load_to_lds pulls mem[hidx[b]] (8 KB) into LDS, wave waits on
//   TENSORcnt, then all 8 waves stream LDS -> d_out (b32, row starts at +1
//   float so 16B stores would be misaligned).
// ---------------------------------------------------------------------------
__global__ __launch_bounds__(256) void gather_kernel(const float* __restrict__ mem,
                                                     const int* __restrict__ hidx,
                                                     float* __restrict__ out) {
    __shared__ float tile[D_];
    const int b = blockIdx.x;
    const int idx = __builtin_amdgcn_readfirstlane(hidx[b]);

    if (threadIdx.x < 32) {
        // LDS byte offset = low 32 bits of the flat shared address
        unsigned lds_off = (unsigned)(uintptr_t)(void*)tile;
        unsigned long long ga = (unsigned long long)(uintptr_t)mem +
                                (unsigned long long)(unsigned)idx * (unsigned long long)(D_ * 4);
        // D# group 0: count=1 | lds_addr | global_addr[56:0] | type=2
        v4u g0;
        g0.x = 1u;
        g0.y = lds_off;
        g0.z = (unsigned)(ga & 0xffffffffull);
        g0.w = (unsigned)((ga >> 32) & 0x01ffffffull) | (2u << 30);
        // D# group 1: data_size=4B, tensor_dim0=D_, tensor_dim1=1, tile=D_x1, stride0=D_
        v8i g1;
        g1[0] = (int)(2u << 16);                          // data_size = 4 bytes
        g1[1] = (int)(((unsigned)D_ & 0xffffu) << 16);    // tensor_dim0[15:0]
        g1[2] = (int)(1u << 16);                          // tensor_dim0 hi=0, tensor_dim1=1
        g1[3] = (int)(((unsigned)D_ & 0xffffu) << 16);    // tensor_dim1 hi=0, tile_dim0=D_
        g1[4] = 1;                                        // tile_dim1 = 1
        g1[5] = D_;                                       // tensor_dim0_stride
        g1[6] = 0;
        g1[7] = 0;
        v4i zz = {0, 0, 0, 0};
#if defined(__clang_major__) && (__clang_major__ >= 23)
        v8i zz8 = {0, 0, 0, 0, 0, 0, 0, 0};
        __builtin_amdgcn_tensor_load_to_lds(g0, g1, zz, zz, zz8, 0);
#else
        __builtin_amdgcn_tensor_load_to_lds(g0, g1, zz, zz, 0);
#endif
        __builtin_amdgcn_s_wait_tensorcnt(0);
    }
    __syncthreads();

    float* __restrict__ orow = out + 1 + (size_t)b * D_;
#pragma unroll
    for (int j = 0; j < D_ / 256; ++j) {
        const int e = j * 256 + threadIdx.x;   // coalesced b32 across the block
        orow[e] = tile[e];
    }
}

// ---------------------------------------------------------------------------
extern "C" void kernel_launch(void* const* d_in, const int* in_sizes, int n_in,
                              void* d_out, int out_size, void* d_ws, size_t ws_size,
                              hipStream_t stream) {
    (void)in_sizes; (void)n_in; (void)out_size; (void)ws_size;
    const float* logits   = (const float*)d_in[0];   // [B, N]
    const float* mem      = (const float*)d_in[1];   // [N, D]
    const int* memCID     = (const int*)d_in[2];     // [N]
    const int* memTID     = (const int*)d_in[3];     // [N]
    const int* camids     = (const int*)d_in[4];     // [B]
    const int* trackids   = (const int*)d_in[5];     // [B]
    float* out            = (float*)d_out;           // [1 + B*D]

    const int P = B_ * SPLITS;                        // 1024 partials
    float* pm = (float*)d_ws;
    float* ps = pm + P;
    float* pn = ps + P;
    float* pp = pn + P;
    float* pv = pp + P;
    int* pi = (int*)(pv + P);
    float* lossb = (float*)(pi + P);
    int* hidx = (int*)(lossb + B_);

    partial_kernel<<<dim3(SPLITS, B_), TPB, 0, stream>>>(
        logits, memCID, memTID, camids, trackids, pm, ps, pn, pp, pv, pi);
    combine_kernel<<<1, B_, 0, stream>>>(pm, ps, pn, pp, pv, pi, lossb, hidx);
    mean_kernel<<<1, 32, 0, stream>>>(lossb, out);
    gather_kernel<<<B_, 256, 0, stream>>>(mem, hidx, out);
}